// QLSTMTagger_65481071404113
// MI455X (gfx1250) — compile-verified
//
#include <hip/hip_runtime.h>
#include <math.h>

#define SEQ    8192
#define EMBED  256
#define HIDDEN 256
#define TAGS   8192
#define GATES  1024   // 4 * HIDDEN
#define INDIM  512    // EMBED + HIDDEN

typedef __attribute__((ext_vector_type(2))) float v2f;
typedef __attribute__((ext_vector_type(8))) float v8f;

// Native fp32 WMMA: D[16x16] = A[16x4] * B[4x16] + C   (V_WMMA_F32_16X16X4_F32)
// 8-arg form: (neg_a, A, neg_b, B, c_mod, C, reuse_a, reuse_b)
__device__ __forceinline__ v8f wmma4(v2f a, v2f b, v8f c) {
    return __builtin_amdgcn_wmma_f32_16x16x4_f32(
        /*neg_a=*/false, a, /*neg_b=*/false, b,
        /*c_mod=*/(short)0, c, /*reuse_a=*/false, /*reuse_b=*/false);
}

// ---------------------------------------------------------------------------
// Kernel 1: xproj[t, g*256+j] = emb[sent[t]] . Wgate[j, 0:256] + bgate[j]
// One wave per block; each wave computes a 16(M=t) x 64(N=gate-out) strip.
// ---------------------------------------------------------------------------
__global__ void __launch_bounds__(32)
xproj_kernel(const int* __restrict__ sent, const float* __restrict__ emb,
             const float* __restrict__ Wf, const float* __restrict__ bf,
             const float* __restrict__ Wi, const float* __restrict__ bi,
             const float* __restrict__ Wg, const float* __restrict__ bg,
             const float* __restrict__ Wo, const float* __restrict__ bo,
             float* __restrict__ xproj)
{
    const int lane = threadIdx.x;
    const int m    = lane & 15;      // M (or N) index within tile
    const int half = lane >> 4;      // which K pair this lane holds
    const int koff = half * 2;

    const int t0 = blockIdx.x * 16;  // time-tile base
    const int n0 = blockIdx.y * 64;  // gate-output column base (0..960)
    const int gate = n0 >> 8;        // 0=f, 1=i, 2=g, 3=o
    const int jb   = n0 & 255;       // column base within the gate

    const float* W;  const float* bias;
    switch (gate) {
        case 0:  W = Wf; bias = bf; break;
        case 1:  W = Wi; bias = bi; break;
        case 2:  W = Wg; bias = bg; break;
        default: W = Wo; bias = bo; break;
    }

    // A: gathered embedding rows. lane's row = sent[t0 + m]
    const float* aptr = emb + (long)sent[t0 + m] * EMBED + koff;
    // B: B[k,n] = W[n,k] (x-part: k in [0,256)), row stride INDIM
    const float* bp0 = W + (long)(jb +  0 + m) * INDIM + koff;
    const float* bp1 = W + (long)(jb + 16 + m) * INDIM + koff;
    const float* bp2 = W + (long)(jb + 32 + m) * INDIM + koff;
    const float* bp3 = W + (long)(jb + 48 + m) * INDIM + koff;

    v8f acc0 = {}, acc1 = {}, acc2 = {}, acc3 = {};
    #pragma unroll 4
    for (int k = 0; k < EMBED; k += 4) {
        v2f a  = *(const v2f*)(aptr + k);
        v2f b0 = *(const v2f*)(bp0 + k);
        v2f b1 = *(const v2f*)(bp1 + k);
        v2f b2 = *(const v2f*)(bp2 + k);
        v2f b3 = *(const v2f*)(bp3 + k);
        acc0 = wmma4(a, b0, acc0);
        acc1 = wmma4(a, b1, acc1);
        acc2 = wmma4(a, b2, acc2);
        acc3 = wmma4(a, b3, acc3);
    }

    // C/D layout: VGPR r holds M = r + 8*half, N = m
    #pragma unroll
    for (int r = 0; r < 8; ++r) {
        const int trow = t0 + r + half * 8;
        float* orow = xproj + (long)trow * GATES + n0;
        orow[ 0 + m] = acc0[r] + bias[jb +  0 + m];
        orow[16 + m] = acc1[r] + bias[jb + 16 + m];
        orow[32 + m] = acc2[r] + bias[jb + 32 + m];
        orow[48 + m] = acc3[r] + bias[jb + 48 + m];
    }
}

// ---------------------------------------------------------------------------
// Kernel 2: serial LSTM recurrence. Single workgroup of 1024 threads:
// thread j computes gate pre-activation j (dot of h with Wh row, 256 MACs),
// threads 0..255 then update c/h. h, c, gates live in LDS; Wh (1 MB total)
// streams from L2 every step.
// ---------------------------------------------------------------------------
__global__ void __launch_bounds__(1024)
lstm_recur_kernel(const float* __restrict__ xproj,
                  const float* __restrict__ Wf, const float* __restrict__ Wi,
                  const float* __restrict__ Wg, const float* __restrict__ Wo,
                  float* __restrict__ h_all)
{
    __shared__ float h[HIDDEN];
    __shared__ float c[HIDDEN];
    __shared__ float gates[GATES];

    const int tid  = threadIdx.x;        // 0..1023
    const int gate = tid >> 8;
    const int row  = tid & 255;

    const float* Wbase = (gate == 0) ? Wf : (gate == 1) ? Wi : (gate == 2) ? Wg : Wo;
    const float* Wh = Wbase + (long)row * INDIM + EMBED;   // h-part columns [256,512)

    if (tid < HIDDEN) { h[tid] = 0.0f; c[tid] = 0.0f; }
    __syncthreads();

    for (int t = 0; t < SEQ; ++t) {
        float acc = xproj[(long)t * GATES + tid];          // x-part + bias
        #pragma unroll 4
        for (int k = 0; k < HIDDEN; k += 4) {
            const float4 w = *(const float4*)(Wh + k);
            const float4 hv = *(const float4*)(&h[k]);
            acc += w.x * hv.x + w.y * hv.y + w.z * hv.z + w.w * hv.w;
        }
        gates[tid] = acc;
        __syncthreads();

        if (tid < HIDDEN) {
            const float f = 1.0f / (1.0f + __expf(-gates[        tid]));
            const float i = 1.0f / (1.0f + __expf(-gates[256 +  tid]));
            const float g = tanhf(gates[512 + tid]);
            const float o = 1.0f / (1.0f + __expf(-gates[768 +  tid]));
            const float cn = f * c[tid] + i * g;
            const float hn = o * tanhf(cn);
            c[tid] = cn;
            h[tid] = hn;
            h_all[(long)t * HIDDEN + tid] = hn;
        }
        __syncthreads();
    }
}

// ---------------------------------------------------------------------------
// Kernel 3: logits[t, v] = h_all[t,:] . W_tag[v,:] + b_tag[v]
// Same WMMA strip structure as kernel 1 (no gather). Operands are L2-resident
// (8 MB + 8 MB); the 256 MB store dominates -> memory bound.
// ---------------------------------------------------------------------------
__global__ void __launch_bounds__(32)
logits_kernel(const float* __restrict__ h_all, const float* __restrict__ Wt,
              const float* __restrict__ bt, float* __restrict__ out)
{
    const int lane = threadIdx.x;
    const int m    = lane & 15;
    const int half = lane >> 4;
    const int koff = half * 2;

    const int t0 = blockIdx.x * 16;
    const int n0 = blockIdx.y * 64;

    const float* aptr = h_all + (long)(t0 + m) * HIDDEN + koff;
    const float* bp0  = Wt + (long)(n0 +  0 + m) * HIDDEN + koff;
    const float* bp1  = Wt + (long)(n0 + 16 + m) * HIDDEN + koff;
    const float* bp2  = Wt + (long)(n0 + 32 + m) * HIDDEN + koff;
    const float* bp3  = Wt + (long)(n0 + 48 + m) * HIDDEN + koff;

    v8f acc0 = {}, acc1 = {}, acc2 = {}, acc3 = {};
    #pragma unroll 4
    for (int k = 0; k < HIDDEN; k += 4) {
        v2f a  = *(const v2f*)(aptr + k);
        v2f b0 = *(const v2f*)(bp0 + k);
        v2f b1 = *(const v2f*)(bp1 + k);
        v2f b2 = *(const v2f*)(bp2 + k);
        v2f b3 = *(const v2f*)(bp3 + k);
        acc0 = wmma4(a, b0, acc0);
        acc1 = wmma4(a, b1, acc1);
        acc2 = wmma4(a, b2, acc2);
        acc3 = wmma4(a, b3, acc3);
    }

    #pragma unroll
    for (int r = 0; r < 8; ++r) {
        const int trow = t0 + r + half * 8;
        float* orow = out + (long)trow * TAGS + n0;
        orow[ 0 + m] = acc0[r] + bt[n0 +  0 + m];
        orow[16 + m] = acc1[r] + bt[n0 + 16 + m];
        orow[32 + m] = acc2[r] + bt[n0 + 32 + m];
        orow[48 + m] = acc3[r] + bt[n0 + 48 + m];
    }
}

// ---------------------------------------------------------------------------
// Kernel 4: in-place row-wise log-softmax over TAGS (one block per row)
// ---------------------------------------------------------------------------
__global__ void __launch_bounds__(256)
logsoftmax_kernel(float* __restrict__ out)
{
    __shared__ float red[256];
    const int tid = threadIdx.x;
    float* row = out + (long)blockIdx.x * TAGS;

    float mx = -INFINITY;
    for (int i = tid; i < TAGS; i += 256) mx = fmaxf(mx, row[i]);
    red[tid] = mx;
    __syncthreads();
    for (int s = 128; s > 0; s >>= 1) {
        if (tid < s) red[tid] = fmaxf(red[tid], red[tid + s]);
        __syncthreads();
    }
    mx = red[0];
    __syncthreads();

    float sum = 0.0f;
    for (int i = tid; i < TAGS; i += 256) sum += __expf(row[i] - mx);
    red[tid] = sum;
    __syncthreads();
    for (int s = 128; s > 0; s >>= 1) {
        if (tid < s) red[tid] += red[tid + s];
        __syncthreads();
    }
    const float lse = mx + __logf(red[0]);

    for (int i = tid; i < TAGS; i += 256) row[i] = row[i] - lse;
}

// ---------------------------------------------------------------------------
extern "C" void kernel_launch(void* const* d_in, const int* in_sizes, int n_in,
                              void* d_out, int out_size, void* d_ws, size_t ws_size,
                              hipStream_t stream)
{
    const int*   sent = (const int*)  d_in[0];
    const float* emb  = (const float*)d_in[1];
    const float* Wf   = (const float*)d_in[2];
    const float* bf   = (const float*)d_in[3];
    const float* Wi   = (const float*)d_in[4];
    const float* bi   = (const float*)d_in[5];
    const float* Wg   = (const float*)d_in[6];
    const float* bg   = (const float*)d_in[7];
    const float* Wo   = (const float*)d_in[8];
    const float* bo   = (const float*)d_in[9];
    const float* Wt   = (const float*)d_in[10];
    const float* bt   = (const float*)d_in[11];
    float* out = (float*)d_out;

    float* xproj = (float*)d_ws;                        // [SEQ, 1024] = 32 MB
    float* h_all = xproj + (size_t)SEQ * GATES;         // [SEQ, 256]  =  8 MB

    dim3 wave(32);

    dim3 g1(SEQ / 16, GATES / 64);
    xproj_kernel<<<g1, wave, 0, stream>>>(sent, emb, Wf, bf, Wi, bi, Wg, bg, Wo, bo, xproj);

    lstm_recur_kernel<<<1, 1024, 0, stream>>>(xproj, Wf, Wi, Wg, Wo, h_all);

    dim3 g3(SEQ / 16, TAGS / 64);
    logits_kernel<<<g3, wave, 0, stream>>>(h_all, Wt, bt, out);

    logsoftmax_kernel<<<SEQ, 256, 0, stream>>>(out);
}